// BatchHardTripletLoss_10565619548445
// MI455X (gfx1250) — compile-verified
//
#include <hip/hip_runtime.h>

// ---------------------------------------------------------------------------
// Batch-hard triplet loss, fused Gram-matrix + row reductions for gfx1250.
// Heavy path: V_WMMA_F32_16X16X4_F32 (fp32 matrix core). The 64x1024 A-block
// lives in the 320KB WGP LDS for the whole kernel; B tiles are staged per
// 128-deep K-chunk, shared by all 8 waves. Latency hiding comes from
// register-free GLOBAL_PREFETCH_B8 (__builtin_prefetch) of the next chunk,
// issued under the WMMA K-loop — no counters, no VGPR cost, so no spills.
// Wave pairs (w, w+4) own the same 16 rows and split the 64-column slab.
// ---------------------------------------------------------------------------

typedef float v2f __attribute__((ext_vector_type(2)));
typedef float v8f __attribute__((ext_vector_type(8)));

constexpr int   kN     = 8192;
constexpr int   kD     = 1024;
constexpr float kALPHA = 0.1f;
constexpr float kEPS   = 1e-7f;
constexpr float kFMAX  = 3.402823466e+38f;

constexpr int kRowsPerWG   = 64;   // 4 row-groups x 16 rows, each owned by 2 waves
constexpr int kColsPerPass = 64;   // each wave covers 2 N-tiles of 16
constexpr int kKC          = 128;  // K chunk staged to LDS for B
// +4 dword padding => consecutive rows shift 4 banks; a 16x4 f32 fragment read
// (float2 per lane, 16 rows per half-wave) then touches all 64 banks once.
constexpr int kAStride = kD + 4;    // 1028 dwords
constexpr int kBStride = kKC + 4;   // 132 dwords
constexpr int kLdsFloats = kRowsPerWG * kAStride + kColsPerPass * kBStride; // 74240
constexpr size_t kLdsBytes = (size_t)kLdsFloats * 4;                        // 296960 < 320KB

// ------------------------------ kernel 1: x_norm ---------------------------
__global__ void __launch_bounds__(256)
xnorm_kernel(const float* __restrict__ H, float* __restrict__ xn) {
    const int row  = blockIdx.x * 8 + (threadIdx.x >> 5);
    const int lane = threadIdx.x & 31;
    const float4* p = (const float4*)(H + (size_t)row * kD);
    float s = 0.0f;
#pragma unroll
    for (int i = 0; i < 8; ++i) {
        float4 v = p[lane + 32 * i];
        s += v.x * v.x + v.y * v.y + v.z * v.z + v.w * v.w;
    }
#pragma unroll
    for (int m = 16; m >= 1; m >>= 1) s += __shfl_xor(s, m, 32);
    if (lane == 0) xn[row] = s;
}

// --------------------- kernel 2: fused Gram + hard mining ------------------
__global__ void __launch_bounds__(256)
gram_hard_kernel(const float* __restrict__ H,
                 const float* __restrict__ xn,
                 const int*   __restrict__ labels,
                 float*       __restrict__ loss) {
    extern __shared__ float lds[];
    float* ldsA = lds;                           // [64][kAStride]
    float* ldsB = lds + kRowsPerWG * kAStride;   // [64][kBStride]

    const int tid      = threadIdx.x;    // 256 threads = 8 waves
    const int wave     = tid >> 5;
    const int lane     = tid & 31;
    const int lane16   = lane & 15;
    const int hi       = lane >> 4;      // 0 = lanes 0-15, 1 = lanes 16-31
    const int pair     = wave & 3;       // row group 0..3 (16 rows each)
    const int colHalf  = wave >> 2;      // 0: cols 0-31 of slab, 1: cols 32-63
    const int rowBase  = blockIdx.x * kRowsPerWG;

    // ---- stage the whole 64 x 1024 A block into LDS (once, 257KB) ----
    {
        const int r = tid >> 2, q = tid & 3;      // 4 threads per row
        const float4* src = (const float4*)(H + (size_t)(rowBase + r) * kD) + q * 64;
        float* dst = &ldsA[r * kAStride + q * 256];
        for (int i = 0; i < 64; ++i) {
            *(float4*)(dst + 4 * i) = src[i];
        }
    }

    // ---- per-row constants for this wave (C-tile row mapping) ----
    const int giBase = rowBase + pair * 16 + hi * 8;   // rows giBase..giBase+7
    float xnr[8];
    int   labr[8];
#pragma unroll
    for (int r = 0; r < 8; ++r) {
        xnr[r]  = xn[giBase + r];
        labr[r] = labels[giBase + r];
    }

    // reduction state per owned row: hardest-positive max, two smallest
    // different-class distances (values only: the 2nd order statistic is
    // invariant to top_k tie-breaking, and the reference reads back the value)
    float hp[8], v1[8], v2[8];
#pragma unroll
    for (int r = 0; r < 8; ++r) { hp[r] = 0.0f; v1[r] = kFMAX; v2[r] = kFMAX; }

    __syncthreads();   // A block visible

    const int aRow  = pair * 16 + lane16;  // A fragment: M = lane%16
    const int kHalf = hi << 1;             // K sub-offset 0 / 2
    const int bRow0 = (colHalf * 2 + 0) * 16 + lane16;
    const int bRow1 = (colHalf * 2 + 1) * 16 + lane16;

    for (int colBase = 0; colBase < kN; colBase += kColsPerPass) {
        v8f acc0 = {}, acc1 = {};   // two 16x16 fp32 tiles per wave

        for (int kb = 0; kb < kD; kb += kKC) {
            __syncthreads();   // previous chunk's fragment reads done
            // ---- stage B chunk: 64 cols x kKC, 4 threads per column ----
            {
                const float4* src =
                    (const float4*)(H + (size_t)(colBase + (tid >> 2)) * kD + kb)
                    + (tid & 3) * 8;
                float* dst = &ldsB[(tid >> 2) * kBStride + (tid & 3) * 32];
#pragma unroll
                for (int i = 0; i < 8; ++i) {
                    float4 v = src[i];
                    *(float4*)(dst + 4 * i) = v;
                }
            }
            __syncthreads();

            // ---- register-free prefetch of the next chunk (wraps to the
            //      next pass); global_prefetch_b8 tracks no counters, so it
            //      rides for free under the WMMA K-loop below ----
            {
                int nkb = kb + kKC, ncol = colBase;
                if (nkb == kD) { nkb = 0; ncol += kColsPerPass; }
                if (ncol < kN) {
                    __builtin_prefetch(
                        H + (size_t)(ncol + (tid >> 2)) * kD + nkb + (tid & 3) * 32,
                        0, 3);
                }
            }

            // ---- K loop: one A fragment feeds 2 WMMAs ----
#pragma unroll 4
            for (int k = 0; k < kKC; k += 4) {
                v2f a  = *(const v2f*)&ldsA[aRow * kAStride + kb + k + kHalf];
                v2f b0 = *(const v2f*)&ldsB[bRow0 * kBStride + k + kHalf];
                acc0 = __builtin_amdgcn_wmma_f32_16x16x4_f32(
                    false, a, false, b0, (short)0, acc0, false, false);
                v2f b1 = *(const v2f*)&ldsB[bRow1 * kBStride + k + kHalf];
                acc1 = __builtin_amdgcn_wmma_f32_16x16x4_f32(
                    false, a, false, b1, (short)0, acc1, false, false);
            }
        }

        // ---- branchless epilogue: distances + hard mining ----
        auto mine = [&](const v8f& a, int tIdx) {
            const int   gj = colBase + (colHalf * 2 + tIdx) * 16 + lane16;
            const float xc = xn[gj];
            const int   lc = labels[gj];
#pragma unroll
            for (int r = 0; r < 8; ++r) {
                // fmaxf(NaN, eps) == eps — reproduces where(isnan,0) + clamps
                float d = fmaxf(fmaf(-2.0f, a[r], xnr[r] + xc), kEPS);
                const bool same = (lc == labr[r]);
                const float dp = (same && gj != (giBase + r)) ? d : 0.0f;
                hp[r] = fmaxf(hp[r], dp);
                const float dn = same ? kFMAX : d;
                v2[r] = fminf(v2[r], fmaxf(v1[r], dn));
                v1[r] = fminf(v1[r], dn);
            }
        };
        mine(acc0, 0);
        mine(acc1, 1);
    }

    // ---- combine across the 16 lanes holding each row's columns ----
#pragma unroll
    for (int m = 8; m >= 1; m >>= 1) {
#pragma unroll
        for (int r = 0; r < 8; ++r) {
            hp[r] = fmaxf(hp[r], __shfl_xor(hp[r], m, 16));
            const float ov1 = __shfl_xor(v1[r], m, 16);
            const float ov2 = __shfl_xor(v2[r], m, 16);
            v2[r] = fminf(fminf(v2[r], ov2), fmaxf(v1[r], ov1));
            v1[r] = fminf(v1[r], ov1);
        }
    }

    // ---- merge wave pairs (column halves) through LDS, emit per-row loss ----
    float* hpA = ldsB;          // reuse B buffer: 3 x 128 floats
    float* v1A = ldsB + 128;
    float* v2A = ldsB + 256;
    __syncthreads();            // done reading ldsB as B tiles
    if (lane16 == 0) {
#pragma unroll
        for (int r = 0; r < 8; ++r) {
            const int idx = wave * 16 + hi * 8 + r;
            hpA[idx] = hp[r]; v1A[idx] = v1[r]; v2A[idx] = v2[r];
        }
    }
    __syncthreads();
    if (tid < kRowsPerWG) {
        const int ia = (tid >> 4) * 16 + (tid & 15);       // colHalf 0
        const int ib = ia + 64;                            // colHalf 1
        const float h  = fmaxf(hpA[ia], hpA[ib]);
        const float a1 = v1A[ia], b1 = v1A[ib];
        const float s2 = fminf(fminf(v2A[ia], v2A[ib]), fmaxf(a1, b1));
        loss[rowBase + tid] = fmaxf(h - s2 + kALPHA, 0.0f);
    }
}

// ------------------------ kernel 3: masked mean ----------------------------
__global__ void __launch_bounds__(256)
reduce_loss_kernel(const float* __restrict__ loss, float* __restrict__ out) {
    __shared__ float ssum[256];
    __shared__ float scnt[256];
    const int tid = threadIdx.x;
    float s = 0.0f, c = 0.0f;
    for (int i = tid; i < kN; i += 256) {
        float l = loss[i];
        if (l > kEPS) { s += l; c += 1.0f; }
    }
    ssum[tid] = s; scnt[tid] = c;
    __syncthreads();
    for (int off = 128; off > 0; off >>= 1) {
        if (tid < off) { ssum[tid] += ssum[tid + off]; scnt[tid] += scnt[tid + off]; }
        __syncthreads();
    }
    if (tid == 0) out[0] = ssum[0] / scnt[0];
}

// ---------------------------------------------------------------------------
extern "C" void kernel_launch(void* const* d_in, const int* in_sizes, int n_in,
                              void* d_out, int out_size, void* d_ws, size_t ws_size,
                              hipStream_t stream) {
    (void)in_sizes; (void)n_in; (void)out_size; (void)ws_size;
    const float* H      = (const float*)d_in[0];
    const int*   labels = (const int*)d_in[1];
    float*       out    = (float*)d_out;

    float* xn   = (float*)d_ws;        // 8192 floats
    float* loss = xn + kN;             // 8192 floats  (64KB workspace total)

    xnorm_kernel<<<dim3(kN / 8), dim3(256), 0, stream>>>(H, xn);
    gram_hard_kernel<<<dim3(kN / kRowsPerWG), dim3(256), kLdsBytes, stream>>>(
        H, xn, labels, loss);
    reduce_loss_kernel<<<dim3(1), dim3(256), 0, stream>>>(loss, out);
}